// TGVDenoiser_37580963840015
// MI455X (gfx1250) — compile-verified
//
#include <hip/hip_runtime.h>
#include <stdint.h>

// ---------------- problem constants ----------------
#define IMH 512
#define IMW 512
#define NB 4
#define NPIX (IMH * IMW)          // 262144 per batch plane
#define TAUC 0.01f
#define RHOC 1.99f
#define SIGMAC (1.0f / (0.01f * 72.0f))
#define NIT 10

// ---------------- tiling ----------------
#define TILE 32
#define HALO 4
#define PAD (TILE + 2 * HALO)     // 40
#define PLANE (PAD * PAD)         // 1600 floats
#define NPLANES 13
#define BLOCK 256

// LDS plane indices
// 0..3: u2 ch0..3   4: x2   5: y   6,7: r2 ch0,1
// 8,9: tmp (later reused as v)   10: x   11,12: r ch0,1
#define SM(pl, i, j) sm[(pl) * PLANE + (i) * PAD + (j)]

typedef uint32_t u32x4 __attribute__((ext_vector_type(4)));
typedef uint32_t u32x8 __attribute__((ext_vector_type(8)));

// Issue one TDM 2D strided-tile load (global -> LDS) using iterate mode:
// one row (w elements) per iteration, `rows` iterations, LDS pitch = PAD
// elements, global pitch = IMW elements. Descriptor per CDNA5 ISA ch.8.
__device__ __forceinline__ void tdm_load_tile(uint32_t lds_byte_addr,
                                              const float* gptr,
                                              uint32_t w, uint32_t rows) {
  uint64_t ga = (uint64_t)(uintptr_t)gptr;

  u32x4 g0;
  g0.x = 1u;                                     // count=1, user descriptor
  g0.y = lds_byte_addr;                          // lds_addr (bytes)
  g0.z = (uint32_t)ga;                           // global_addr[31:0]
  g0.w = ((uint32_t)(ga >> 32) & 0x01FFFFFFu)    // global_addr[56:32]
         | 0x80000000u;                          // type = 2 ("image")

  u32x8 g1;
  g1.s0 = (2u << 16)                             // data_size = 4B
          | (1u << 19);                          // iterate_enable
  g1.s1 = (w & 0xFFFFu) << 16;                   // tensor_dim0[15:0]
  g1.s2 = (w >> 16) | (1u << 16);                // dim0 hi | tensor_dim1 = 1
  g1.s3 = (w & 0xFFFFu) << 16;                   // dim1 hi(0) | tile_dim0 = w
  g1.s4 = 1u;                                    // tile_dim1 = 1, tile_dim2 = 0
  g1.s5 = (uint32_t)IMW;                         // tensor_dim0_stride lo
  g1.s6 = 0u;
  g1.s7 = 0u;

  u32x4 g2;
  g2.x = 1u;                                     // tensor_dim2 (unused)
  g2.y = (uint32_t)PAD;                          // lds_addr_increment (elems)
  g2.z = (uint32_t)IMW;                          // global_addr_increment lo
  g2.w = ((rows - 1u) & 0xFFFFu) << 16;          // iterate_count (0 => 1x)

  u32x4 g3;
  g3.x = 0u; g3.y = 0u; g3.z = 0u; g3.w = 0u;

  asm volatile("tensor_load_to_lds %0, %1, %2, %3"
               :
               : "s"(g0), "s"(g1), "s"(g2), "s"(g3)
               : "memory");
}

__global__ __launch_bounds__(BLOCK) void tgv_iter_kernel(
    const float* __restrict__ y, const int* __restrict__ ths,
    const float* __restrict__ x2i, const float* __restrict__ r2i,
    const float* __restrict__ u2i, float* __restrict__ x2o,
    float* __restrict__ r2o, float* __restrict__ u2o) {
  __shared__ float sm[NPLANES * PLANE];

  const int b = blockIdx.z;
  const int oi = blockIdx.y * TILE;
  const int oj = blockIdx.x * TILE;
  const int tid = threadIdx.x;

  // ---- clamped tile geometry (uniform per workgroup) ----
  const int x0 = oj - HALO, y0 = oi - HALO;
  const int gx0 = x0 < 0 ? 0 : x0;
  const int gy0 = y0 < 0 ? 0 : y0;
  const int sl = gx0 - x0, st = gy0 - y0;                // 0..4 skipped cols/rows
  int w = PAD - sl;  { int rem = IMW - gx0; if (rem < w) w = rem; }
  int h = PAD - st;  { int rem = IMH - gy0; if (rem < h) h = rem; }
  const uint32_t ldsoff = (uint32_t)(st * PAD + sl) * 4u;
  const size_t gbase = (size_t)b * NPIX + (size_t)gy0 * IMW + gx0;

  // ---- wave 0 issues 8 TDM tile loads, waits on TENSORcnt ----
  if (tid < 32) {
    const float* src[8];
    src[0] = u2i + (size_t)0 * NB * NPIX;
    src[1] = u2i + (size_t)1 * NB * NPIX;
    src[2] = u2i + (size_t)2 * NB * NPIX;
    src[3] = u2i + (size_t)3 * NB * NPIX;
    src[4] = x2i;
    src[5] = y;
    src[6] = r2i + (size_t)0 * NB * NPIX;
    src[7] = r2i + (size_t)1 * NB * NPIX;
#pragma unroll
    for (int p = 0; p < 8; ++p) {
      uint32_t lds = (uint32_t)(uintptr_t)(&sm[p * PLANE]) + ldsoff;
      tdm_load_tile(lds, src[p] + gbase, (uint32_t)w, (uint32_t)h);
    }
    __builtin_amdgcn_s_wait_tensorcnt(0);
  }
  __syncthreads();

  const float thsf = (float)ths[0];
  const float lam1 = 0.1f * thsf;
  const float lam2 = 0.15f * thsf;
  const float taulam1 = TAUC * lam1;
  const float inv1ptau = 1.0f / (1.0f + TAUC);

  // ---- stage A: tmp = TAU * epsilon2_adjoint(u2) on halo-3 region ----
  for (int idx = tid; idx < PLANE; idx += BLOCK) {
    const int pi = idx / PAD, pj = idx % PAD;
    if (pi >= 1 && pi < PAD - 1 && pj >= 1 && pj < PAD - 1) {
      const int gi = oi + pi - HALO, gj = oj + pj - HALO;
      float t0 = 0.0f, t1 = 0.0f;
      if (gi >= 0 && gi < IMH && gj >= 0 && gj < IMW) {
        t0 = SM(0, pi, pj)
             - (gi < IMH - 1 ? SM(0, pi + 1, pj) : 0.0f)
             - (gj < IMW - 1 ? SM(1, pi, pj + 1) : 0.0f)
             + (gj >= 1      ? SM(1, pi, pj)     : 0.0f);
        t1 = SM(2, pi, pj)
             - (gj < IMW - 1 ? SM(2, pi, pj + 1) : 0.0f)
             - (gi < IMH - 1 ? SM(3, pi, pj)     : 0.0f)
             + (gi >= 1      ? SM(3, pi - 1, pj) : 0.0f);
        t0 *= TAUC;
        t1 *= TAUC;
      }
      SM(8, pi, pj) = t0;
      SM(9, pi, pj) = t1;
    }
  }
  __syncthreads();

  // ---- stage B: x (prox_tau_fx) and r (prox_tau_fr) on halo-2 region ----
  for (int idx = tid; idx < PLANE; idx += BLOCK) {
    const int pi = idx / PAD, pj = idx % PAD;
    if (pi >= 2 && pi < PAD - 2 && pj >= 2 && pj < PAD - 2) {
      const int gi = oi + pi - HALO, gj = oj + pj - HALO;
      float xv = 0.0f, r0 = 0.0f, r1 = 0.0f;
      if (gi >= 0 && gi < IMH && gj >= 0 && gj < IMW) {
        float adj = 0.0f;
        adj -= (gj <= IMW - 2 ? SM(8, pi, pj)     : 0.0f);
        adj += (gj >= 1       ? SM(8, pi, pj - 1) : 0.0f);
        adj -= (gi <= IMH - 2 ? SM(9, pi, pj)     : 0.0f);
        adj += (gi >= 1       ? SM(9, pi - 1, pj) : 0.0f);
        xv = (SM(4, pi, pj) - adj + TAUC * SM(5, pi, pj)) * inv1ptau;

        const float rr0 = SM(6, pi, pj) + SM(8, pi, pj);
        const float rr1 = SM(7, pi, pj) + SM(9, pi, pj);
        const float n = sqrtf(rr0 * rr0 + rr1 * rr1);
        const float m = fmaxf(n / taulam1, 1.0f);
        r0 = rr0 - rr0 / m;
        r1 = rr1 - rr1 / m;
      }
      SM(10, pi, pj) = xv;
      SM(11, pi, pj) = r0;
      SM(12, pi, pj) = r1;
    }
  }
  __syncthreads();

  // ---- stage C: v = nabla2(2x - x2) - (2r - r2), halo-1, reuse planes 8/9 ----
  for (int idx = tid; idx < PLANE; idx += BLOCK) {
    const int pi = idx / PAD, pj = idx % PAD;
    if (pi >= 3 && pi < PAD - 3 && pj >= 3 && pj < PAD - 3) {
      const int gi = oi + pi - HALO, gj = oj + pj - HALO;
      float v0 = 0.0f, v1 = 0.0f;
      if (gi >= 0 && gi < IMH && gj >= 0 && gj < IMW) {
        const float xbc = 2.0f * SM(10, pi, pj) - SM(4, pi, pj);
        const float d0 = (gj <= IMW - 2)
            ? (2.0f * SM(10, pi, pj + 1) - SM(4, pi, pj + 1)) - xbc : 0.0f;
        const float d1 = (gi <= IMH - 2)
            ? (2.0f * SM(10, pi + 1, pj) - SM(4, pi + 1, pj)) - xbc : 0.0f;
        v0 = d0 - (2.0f * SM(11, pi, pj) - SM(6, pi, pj));
        v1 = d1 - (2.0f * SM(12, pi, pj) - SM(7, pi, pj));
      }
      SM(8, pi, pj) = v0;
      SM(9, pi, pj) = v1;
    }
  }
  __syncthreads();

  // ---- stage D: u = prox_sigma_g_conj(u2 + SIGMA*epsilon2(v)); over-relax ----
#pragma unroll
  for (int k = 0; k < 4; ++k) {
    const int r = (tid >> 5) + 8 * k;
    const int c = tid & 31;
    const int pi = HALO + r, pj = HALO + c;
    const int gi = oi + r, gj = oj + c;

    const float v0c = SM(8, pi, pj), v1c = SM(9, pi, pj);
    const float G0 = v0c - (gi >= 1 ? SM(8, pi - 1, pj) : 0.0f);
    const float G1 = (gj >= 1) ? (v0c - SM(8, pi, pj - 1)) : 0.0f;
    const float G2 = v1c - (gj >= 1 ? SM(9, pi, pj - 1) : 0.0f);
    const float G3 = (gi <= IMH - 2) ? (SM(9, pi + 1, pj) - v1c) : 0.0f;

    const float a0 = SM(0, pi, pj), a1 = SM(1, pi, pj);
    const float a2 = SM(2, pi, pj), a3 = SM(3, pi, pj);
    float u0 = a0 + SIGMAC * G0;
    float u1 = a1 + SIGMAC * G1;
    float u2 = a2 + SIGMAC * G2;
    float u3 = a3 + SIGMAC * G3;
    const float nrm = sqrtf(u0 * u0 + u1 * u1 + u2 * u2 + u3 * u3);
    const float inv = 1.0f / fmaxf(nrm / lam2, 1.0f);
    u0 *= inv; u1 *= inv; u2 *= inv; u3 *= inv;

    const size_t pix = (size_t)b * NPIX + (size_t)gi * IMW + gj;

    const float x2v = SM(4, pi, pj);
    x2o[pix] = x2v + RHOC * (SM(10, pi, pj) - x2v);

    const float r20 = SM(6, pi, pj), r21 = SM(7, pi, pj);
    r2o[pix]                     = r20 + RHOC * (SM(11, pi, pj) - r20);
    r2o[(size_t)NB * NPIX + pix] = r21 + RHOC * (SM(12, pi, pj) - r21);

    u2o[(size_t)0 * NB * NPIX + pix] = a0 + RHOC * (u0 - a0);
    u2o[(size_t)1 * NB * NPIX + pix] = a1 + RHOC * (u1 - a1);
    u2o[(size_t)2 * NB * NPIX + pix] = a2 + RHOC * (u2 - a2);
    u2o[(size_t)3 * NB * NPIX + pix] = a3 + RHOC * (u3 - a3);
  }
}

__global__ __launch_bounds__(BLOCK) void tgv_init_kernel(
    const float* __restrict__ y, float* __restrict__ x2,
    float* __restrict__ r2, float* __restrict__ u2) {
  const size_t N = (size_t)NB * NPIX;
  const size_t t = (size_t)blockIdx.x * BLOCK + threadIdx.x;
  if (t < N) {
    x2[t] = y[t];
    r2[t] = 0.0f;
    r2[N + t] = 0.0f;
    u2[t] = 0.0f;
    u2[N + t] = 0.0f;
    u2[2 * N + t] = 0.0f;
    u2[3 * N + t] = 0.0f;
  }
}

extern "C" void kernel_launch(void* const* d_in, const int* in_sizes, int n_in,
                              void* d_out, int out_size, void* d_ws,
                              size_t ws_size, hipStream_t stream) {
  (void)in_sizes; (void)n_in; (void)out_size; (void)ws_size;
  const float* y = (const float*)d_in[0];
  const int* ths = (const int*)d_in[1];

  const size_t N = (size_t)NB * NPIX;       // 1,048,576 floats per field plane-set
  float* ws = (float*)d_ws;                 // 13*N floats = 52 MiB used
  float* x2B = ws;                          // N
  float* r2A = ws + N;                      // 2N (planar: [ch][b][h][w])
  float* r2B = ws + 3 * N;                  // 2N
  float* u2A = ws + 5 * N;                  // 4N
  float* u2B = ws + 9 * N;                  // 4N
  float* x2A = (float*)d_out;               // iteration 9 (odd) writes into A

  tgv_init_kernel<<<(unsigned)((N + BLOCK - 1) / BLOCK), BLOCK, 0, stream>>>(
      y, x2A, r2A, u2A);

  const dim3 grid(IMW / TILE, IMH / TILE, NB);
  for (int it = 0; it < NIT; ++it) {
    const bool even = (it & 1) == 0;
    tgv_iter_kernel<<<grid, BLOCK, 0, stream>>>(
        y, ths,
        even ? x2A : x2B, even ? r2A : r2B, even ? u2A : u2B,
        even ? x2B : x2A, even ? r2B : r2A, even ? u2B : u2A);
  }
}